// NTMMemory_14001593385379
// MI455X (gfx1250) — compile-verified
//
#include <hip/hip_runtime.h>
#include <hip/hip_bf16.h>
#include <math.h>

// Shapes from the reference
#define BB   128
#define NN   2048
#define MM   128
#define OFF  1e-6f
#define CEPS 1e-8f
#define OUTROW (MM + 2*NN)   // 4224

typedef __attribute__((ext_vector_type(2))) float v2f;
typedef __attribute__((ext_vector_type(8))) float v8f;

// d_ws layout (floats): S[B*M] | Z[B] | P[B] | ebuf[B*N]
#define WS_S     0
#define WS_Z     (BB*MM)
#define WS_P     (BB*MM + BB)
#define WS_EBUF  (BB*MM + 2*BB)
#define WS_NZERO (BB*MM + 2*BB)        // floats to zero (S,Z,P)

__global__ void ntm_zero(float* __restrict__ ws, int n) {
    int i = blockIdx.x * blockDim.x + threadIdx.x;
    if (i < n) ws[i] = 0.0f;
}

// Main pass: one block = 128 threads (4 waves) = 64 rows of one batch.
// grid = (N/64, B)
__global__ void __launch_bounds__(128)
ntm_pass1(const float* __restrict__ memory,
          const float* __restrict__ k_t,
          const float* __restrict__ beta_t,
          const float* __restrict__ gamma_t,
          const float* __restrict__ w_prev,
          float* __restrict__ S,      // [B*M]
          float* __restrict__ Z,      // [B]
          float* __restrict__ P,      // [B]
          float* __restrict__ ebuf)   // [B*N]
{
    const int b    = blockIdx.y;
    const int n0   = blockIdx.x * 64;
    const int t    = threadIdx.x;      // 0..127
    const int lane = t & 31;
    const int wave = t >> 5;

    __shared__ float tile[64 * 132];   // 64 rows, padded stride 132 (bank-conflict-free)
    __shared__ float kbuf[MM];         // key + OFFSET
    __shared__ float dotbuf[64];
    __shared__ float nmbuf[64];
    __shared__ float sred[4];          // [0]=nk2  [1]=sum exp  [2]=sum w^gamma

    if (t < 4) sred[t] = 0.0f;
    kbuf[t] = k_t[b * MM + t] + OFF;
    __syncthreads();

    // ---- nk^2 = sum (k+off)^2 : wave reduce + LDS atomic ----
    {
        float v = kbuf[t]; v *= v;
        for (int o = 16; o; o >>= 1) v += __shfl_xor(v, o, 32);
        if (lane == 0) atomicAdd(&sred[0], v);
    }

    // ---- stage 64x128 memory tile into LDS (coalesced float4, +OFFSET) ----
    const float* memb = memory + ((size_t)b * NN + n0) * MM;
    for (int it = 0; it < 16; ++it) {
        const int R = it * 4 + wave;                       // one full row per wave
        float4 v = *(const float4*)(memb + (size_t)R * MM + 4 * lane);
        v.x += OFF; v.y += OFF; v.z += OFF; v.w += OFF;
        *(float4*)&tile[R * 132 + 4 * lane] = v;
    }
    __syncthreads();

    const float nk = fmaxf(sqrtf(sred[0]), CEPS);

    // ---- WMMA f32 16x16x4: dot[n] = sum_m mk[n,m] * kk[m] for this wave's 16 rows ----
    // A layout (16x4 f32): lane L holds row L%16; K-offset 0,1 (lanes<16) / 2,3 (lanes>=16)
    // B = key replicated across the 16 columns -> every column of D is the dot vector.
    v8f acc = {0.f, 0.f, 0.f, 0.f, 0.f, 0.f, 0.f, 0.f};
    const int rowa = lane & 15;
    const int koff = (lane < 16) ? 0 : 2;
    const float* arow = &tile[(wave * 16 + rowa) * 132];
    float nm2 = 0.0f;
    for (int c = 0; c < 32; ++c) {
        v2f a, kb;
        a.x  = arow[4 * c + koff];
        a.y  = arow[4 * c + koff + 1];
        kb.x = kbuf[4 * c + koff];
        kb.y = kbuf[4 * c + koff + 1];
        acc = __builtin_amdgcn_wmma_f32_16x16x4_f32(
                  false, a, false, kb, (short)0, acc, false, false);
        nm2 = fmaf(a.x, a.x, nm2);
        nm2 = fmaf(a.y, a.y, nm2);
    }
    // row norm: halves of each row live in lanes L and L^16
    nm2 += __shfl_xor(nm2, 16, 32);
    if (lane < 16) nmbuf[wave * 16 + lane] = nm2;
    // D layout: lanes 0-15 hold rows 0-7 in acc[0..7]; lanes 16-31 hold rows 8-15
    if (lane == 0) {
        #pragma unroll
        for (int r = 0; r < 8; ++r) dotbuf[wave * 16 + r] = acc[r];
    }
    if (lane == 16) {
        #pragma unroll
        for (int r = 0; r < 8; ++r) dotbuf[wave * 16 + 8 + r] = acc[r];
    }
    __syncthreads();

    // ---- per-row tail: cosine -> exp(beta*K) ; plus w_prev^gamma partials ----
    const float beta  = beta_t[b];
    const float gamma = gamma_t[b];
    if (t < 64) {   // waves 0,1 fully active (wave-uniform)
        const float nm = fmaxf(sqrtf(nmbuf[t]), CEPS);
        const float K  = dotbuf[t] / (nm * nk);
        const float e  = __expf(beta * K);           // |beta*K| <= ~1, no max-sub needed
        ebuf[(size_t)b * NN + n0 + t] = e;
        float wp = __powf(w_prev[(size_t)b * NN + n0 + t], gamma);
        float es = e, ps = wp;
        for (int o = 16; o; o >>= 1) {
            es += __shfl_xor(es, o, 32);
            ps += __shfl_xor(ps, o, 32);
        }
        if (lane == 0) { atomicAdd(&sred[1], es); atomicAdd(&sred[2], ps); }
    }
    __syncthreads();
    if (t == 0) {
        atomicAdd(&Z[b], sred[1]);
        atomicAdd(&P[b], sred[2]);
    }

    // ---- column sums of raw memory over this tile ----
    float cs = 0.0f;
    for (int R = 0; R < 64; ++R) cs += tile[R * 132 + t];
    cs -= 64.0f * OFF;   // undo the +OFFSET baked into the tile
    atomicAdd(&S[b * MM + t], cs);
}

// Finalize: r_t (closed form via column sums), w_t (per-batch constant), w_g.
// grid = B blocks x 256 threads
__global__ void __launch_bounds__(256)
ntm_finalize(const float* __restrict__ S,
             const float* __restrict__ Z,
             const float* __restrict__ P,
             const float* __restrict__ e_t,
             const float* __restrict__ a_t,
             const float* __restrict__ w_prev,
             const float* __restrict__ g_t,
             const float* __restrict__ ebuf,
             float* __restrict__ out)
{
    const int b = blockIdx.x;
    const int t = threadIdx.x;
    const float wsc  = P[b] + OFF;
    const float invZ = 1.0f / Z[b];
    const float g    = g_t[b];
    float* orow = out + (size_t)b * OUTROW;

    if (t < MM) {
        const float s = S[b * MM + t];
        const float r = wsc * ((1.0f - wsc * e_t[b * MM + t]) * s
                               + (float)NN * wsc * a_t[b * MM + t]);
        orow[t] = r;
    }
    for (int n = t; n < NN; n += 256) {
        orow[MM + n] = wsc;                               // broadcast w_t
        const float wc = ebuf[(size_t)b * NN + n] * invZ; // softmax
        orow[MM + NN + n] = g * wc + (1.0f - g) * w_prev[(size_t)b * NN + n];
    }
}

extern "C" void kernel_launch(void* const* d_in, const int* in_sizes, int n_in,
                              void* d_out, int out_size, void* d_ws, size_t ws_size,
                              hipStream_t stream) {
    const float* memory  = (const float*)d_in[0];   // [B,N,M]
    const float* k_t     = (const float*)d_in[1];   // [B,M]
    const float* beta_t  = (const float*)d_in[2];   // [B]
    const float* g_t     = (const float*)d_in[3];   // [B]
    const float* w_prev  = (const float*)d_in[4];   // [B,N]
    /* d_in[5] = s_t : unused by reference */
    const float* gamma_t = (const float*)d_in[6];   // [B]
    const float* e_t     = (const float*)d_in[7];   // [B,M]
    const float* a_t     = (const float*)d_in[8];   // [B,M]

    float* ws   = (float*)d_ws;
    float* S    = ws + WS_S;
    float* Z    = ws + WS_Z;
    float* P    = ws + WS_P;
    float* ebuf = ws + WS_EBUF;
    float* out  = (float*)d_out;

    ntm_zero<<<(WS_NZERO + 255) / 256, 256, 0, stream>>>(ws, WS_NZERO);
    ntm_pass1<<<dim3(NN / 64, BB), 128, 0, stream>>>(
        memory, k_t, beta_t, gamma_t, w_prev, S, Z, P, ebuf);
    ntm_finalize<<<BB, 256, 0, stream>>>(S, Z, P, e_t, a_t, w_prev, g_t, ebuf, out);
}